// BatchAllTtripletLoss_9079560864326
// MI455X (gfx1250) — compile-verified
//
#include <hip/hip_runtime.h>

// Problem constants from the reference
#define B 512
#define D 256
#define NLBL_MARGIN 1.0f
#define EPSF 1e-8f

typedef float v2f __attribute__((ext_vector_type(2)));
typedef float v8f __attribute__((ext_vector_type(8)));

// ---------------------------------------------------------------------------
// Kernel 1: squared norms, one wave (32 lanes) per row
// ---------------------------------------------------------------------------
__global__ void sq_kernel(const float* __restrict__ X, float* __restrict__ sq) {
    int i    = blockIdx.x;        // 0..511
    int lane = threadIdx.x;       // 0..31
    const float* row = X + (size_t)i * D;
    float s = 0.0f;
    #pragma unroll
    for (int c = lane; c < D; c += 32) {
        float v = row[c];
        s += v * v;
    }
    #pragma unroll
    for (int off = 16; off > 0; off >>= 1)
        s += __shfl_down(s, off, 32);
    if (lane == 0) sq[i] = s;
}

// ---------------------------------------------------------------------------
// Kernel 2: distance matrix via V_WMMA_F32_16X16X4_F32.
// One wave computes one 16x16 tile of X * X^T, K = 256 in 64 steps of 4.
// A-layout (16x4 f32): lane l holds A[l%16][2*(l/16)+r] in VGPR r.
// B-layout (4x16 f32): lane l holds B[2*(l/16)+r][l%16] in VGPR r.
// For a Gram matrix both are contiguous float2 loads from rows of X.
// ---------------------------------------------------------------------------
__global__ void dist_kernel(const float* __restrict__ X,
                            const float* __restrict__ sq,
                            float* __restrict__ dist) {
    int gtid = blockIdx.x * blockDim.x + threadIdx.x;
    int wave = gtid >> 5;          // global wave id = tile id, 0..1023
    int lane = threadIdx.x & 31;

    int ti = wave >> 5;            // tile row, 0..31
    int tj = wave & 31;            // tile col, 0..31
    int i0 = ti * 16;
    int j0 = tj * 16;

    int m    = lane & 15;          // row within tile (A) / col within tile (B)
    int half = lane >> 4;          // K-half selector

    const float* arow = X + (size_t)(i0 + m) * D + 2 * half;
    const float* brow = X + (size_t)(j0 + m) * D + 2 * half;

    v8f c = {};
    #pragma unroll 8
    for (int kb = 0; kb < D; kb += 4) {
        v2f a = *(const v2f*)(arow + kb);
        v2f b = *(const v2f*)(brow + kb);
        // (neg_a, A, neg_b, B, c_mod, C, reuse_a, reuse_b)
        c = __builtin_amdgcn_wmma_f32_16x16x4_f32(
                false, a, false, b, (short)0, c, false, false);
    }

    // C/D layout: lane l, VGPR v -> M = v + 8*(l/16), N = l%16
    int n  = lane & 15;
    float sj = sq[j0 + n];
    #pragma unroll
    for (int v = 0; v < 8; ++v) {
        int ii = i0 + v + 8 * half;
        float d2 = sq[ii] + sj - 2.0f * c[v];
        d2 = fmaxf(d2, 0.0f);
        dist[(size_t)ii * B + (j0 + n)] = __builtin_sqrtf(d2);
    }
}

// ---------------------------------------------------------------------------
// Kernel 3: per-anchor triplet partial sums. One wave per anchor i.
// Phase 1: ballot-compact positive distances d(i,j) (same label, j!=i) to LDS.
// Phase 2: lanes sweep k; for negatives accumulate relu(d_ij - d_ik + margin).
// ---------------------------------------------------------------------------
__global__ void triplet_kernel(const float* __restrict__ dist,
                               const int* __restrict__ labels,
                               float* __restrict__ psum,
                               float* __restrict__ pcnt) {
    __shared__ float posd[B];

    int i    = blockIdx.x;        // anchor
    int lane = threadIdx.x;       // 0..31
    int li   = labels[i];
    const float* drow = dist + (size_t)i * B;

    // Phase 1: build compacted positive-distance list in LDS
    int base = 0;
    for (int c = 0; c < B / 32; ++c) {
        int j = c * 32 + lane;
        bool p = (labels[j] == li) && (j != i);
        unsigned msk = __builtin_amdgcn_ballot_w32(p);
        if (p) {
            int idx = base + __popc(msk & ((1u << lane) - 1u));
            posd[idx] = drow[j];
        }
        base += __popc(msk);
    }
    int npos = base;
    __syncthreads();

    // Phase 2: sweep negatives
    float sum = 0.0f, cnt = 0.0f;
    for (int c = 0; c < B / 32; ++c) {
        int k = c * 32 + lane;
        float dk = drow[k];
        if (labels[k] != li) {
            for (int p = 0; p < npos; ++p) {
                float v = posd[p] - dk + NLBL_MARGIN;
                if (v > 0.0f)  sum += v;
                if (v > EPSF)  cnt += 1.0f;
            }
        }
    }

    #pragma unroll
    for (int off = 16; off > 0; off >>= 1) {
        sum += __shfl_down(sum, off, 32);
        cnt += __shfl_down(cnt, off, 32);
    }
    if (lane == 0) { psum[i] = sum; pcnt[i] = cnt; }
}

// ---------------------------------------------------------------------------
// Kernel 4: deterministic fixed-order final reduction
// ---------------------------------------------------------------------------
__global__ void final_kernel(const float* __restrict__ psum,
                             const float* __restrict__ pcnt,
                             float* __restrict__ out) {
    if (threadIdx.x == 0 && blockIdx.x == 0) {
        float S = 0.0f, C = 0.0f;
        for (int i = 0; i < B; ++i) { S += psum[i]; C += pcnt[i]; }
        out[0] = S / (C + EPSF);
    }
}

// ---------------------------------------------------------------------------
extern "C" void kernel_launch(void* const* d_in, const int* in_sizes, int n_in,
                              void* d_out, int out_size, void* d_ws, size_t ws_size,
                              hipStream_t stream) {
    const float* X      = (const float*)d_in[0];   // [512, 256] fp32
    const int*   labels = (const int*)d_in[1];     // [512] int

    float* ws   = (float*)d_ws;
    float* dist = ws;                      // 512*512 floats
    float* sq   = ws + (size_t)B * B;      // 512 floats
    float* psum = sq + B;                  // 512 floats
    float* pcnt = psum + B;                // 512 floats

    sq_kernel<<<B, 32, 0, stream>>>(X, sq);

    // 1024 tiles, 1 wave per tile, 4 waves (128 threads) per block -> 256 blocks
    dist_kernel<<<(B / 16) * (B / 16) / 4, 128, 0, stream>>>(X, sq, dist);

    triplet_kernel<<<B, 32, 0, stream>>>(dist, labels, psum, pcnt);

    final_kernel<<<1, 32, 0, stream>>>(psum, pcnt, (float*)d_out);
}